// IRT_16492674417200
// MI455X (gfx1250) — compile-verified
//
#include <hip/hip_runtime.h>
#include <math.h>

typedef __attribute__((ext_vector_type(2))) float v2f;
typedef __attribute__((ext_vector_type(8))) float v8f;
typedef unsigned int u32x4 __attribute__((ext_vector_type(4)));
typedef int i32x4 __attribute__((ext_vector_type(4)));
typedef int i32x8 __attribute__((ext_vector_type(8)));

#define BPW   8          // batches per workgroup
#define LP    260        // padded LDS row pitch (floats) -> kills stride-256 bank conflicts
#define NER   (BPW*6)    // 48 E rows per WG
#define NKR   (BPW*8)    // 64 K rows per WG

// LDS float offsets
#define OFF_E    0
#define OFF_K    (OFF_E   + NER*LP)
#define OFF_LE   (OFF_K   + NKR*LP)
#define OFF_LK   (OFF_LE  + NER*LP)
#define OFF_DAN  (OFF_LK  + NKR*LP)
#define OFF_SIG  (OFF_DAN + BPW*LP)
#define OFF_ISIG (OFF_SIG + 4*LP)
#define OFF_CO   (OFF_ISIG + 4)
#define OFF_TOL  (OFF_CO  + NER)
#define OFF_PSS  (OFF_TOL + NER)
#define OFF_C    (OFF_PSS + NKR)
#define SMEM_FLOATS (OFF_C + BPW*48)       // 61908 floats
#define SMEM_BYTES  (SMEM_FLOATS * 4)      // 247632 B < 320KB/WGP

__device__ __forceinline__ float wred32(float x) {
    #pragma unroll
    for (int o = 16; o; o >>= 1) x += __shfl_down(x, o, 32);
    return x;
}
__device__ __forceinline__ float g8sum(float x) {
    #pragma unroll
    for (int o = 4; o; o >>= 1) x += __shfl_xor(x, o, 8);
    return x;
}
__device__ __forceinline__ float g8max(float x) {
    #pragma unroll
    for (int o = 4; o; o >>= 1) x = fmaxf(x, __shfl_xor(x, o, 8));
    return x;
}

// Issue one TDM 2D tile load: global (rows x 256 f32, stride 256) -> LDS at
// lds_byte, with 4-DWORD padding after every 256 DWORDs (=> 260-float pitch).
// Toolchain is clang-23 (therock): 6-arg builtin
// (uint32x4 g0, int32x8 g1, int32x4, int32x4, int32x8, i32 cpol).
__device__ __forceinline__ void tdm_load_tile(const float* gsrc, unsigned lds_byte, int rows) {
    unsigned long long ga = (unsigned long long)(uintptr_t)gsrc;
    u32x4 g0;
    g0[0] = 1u;                                            // count=1, is_restore=0, gather off
    g0[1] = lds_byte;                                      // lds_addr (bytes)
    g0[2] = (unsigned)(ga & 0xFFFFFFFFu);                  // global_addr[31:0]
    g0[3] = (unsigned)((ga >> 32) & 0x01FFFFFFu) | (2u << 30);  // addr[56:32] | type=2
    i32x8 g1;
    g1[0] = (int)((2u << 16)      // data_size = 4B
                | (1u << 20)      // pad_enable
                | (7u << 22)      // pad_interval: 256 DWORDs
                | (3u << 25));    // pad_amount: 4 DWORDs
    g1[1] = (int)(256u << 16);    // tensor_dim0 = 256 (low16 in [63:48])
    g1[2] = (int)((unsigned)rows << 16);   // tensor_dim0 hi=0 | tensor_dim1 = rows
    g1[3] = (int)(256u << 16);    // tensor_dim1 hi=0 | tile_dim0 = 256
    g1[4] = rows;                 // tile_dim1 = rows | tile_dim2 = 0
    g1[5] = 256;                  // tensor_dim0_stride = 256 elements
    g1[6] = 0;
    g1[7] = 0;
    i32x4 gz4 = {0, 0, 0, 0};     // groups 2/3 unused (tile_dim2==0)
    i32x8 gz8 = {0, 0, 0, 0, 0, 0, 0, 0};
    __builtin_amdgcn_tensor_load_to_lds(g0, g1, gz4, gz4, gz8, 0);
}

__global__ __launch_bounds__(256, 1)
void irt_fused(const float* __restrict__ E, const float* __restrict__ K,
               const float* __restrict__ danger, const float* __restrict__ w_prev,
               const float* __restrict__ fitness, const float* __restrict__ crisis_level,
               const float* __restrict__ delta_sharpe, const float* __restrict__ Lmat,
               const float* __restrict__ sigs, float* __restrict__ out)
{
    extern __shared__ float smem[];
    float* sE   = smem + OFF_E;
    float* sK   = smem + OFF_K;
    float* sLE  = smem + OFF_LE;
    float* sLK  = smem + OFF_LK;
    float* sDan = smem + OFF_DAN;
    float* sSig = smem + OFF_SIG;
    float* sIsg = smem + OFF_ISIG;
    float* sCo  = smem + OFF_CO;
    float* sTol = smem + OFF_TOL;
    float* sPss = smem + OFF_PSS;
    float* sC   = smem + OFF_C;

    const int tid  = threadIdx.x;
    const int wave = tid >> 5, lane = tid & 31;
    const int b0   = blockIdx.x * BPW;

    // ---------------- Phase 0: stage via TDM (E, K) + vector loads (danger, sigs) ----------------
    if (wave == 0) {
        unsigned base = (unsigned)(uintptr_t)smem;   // LDS aperture: low 32 bits = LDS byte offset
        tdm_load_tile(E + (size_t)(b0 * 6) * 256, base + OFF_E * 4, NER);
        tdm_load_tile(K + (size_t)(b0 * 8) * 256, base + OFF_K * 4, NKR);
    }
    // pull the shared 256KB metric into L2 early
    __builtin_prefetch(Lmat + (size_t)tid * 256, 0, 1);

    for (int idx = tid; idx < BPW * 64; idx += 256) {        // danger: 8 rows x 64 float4
        int r = idx >> 6, c4 = (idx & 63) * 4;
        *(float4*)&sDan[r * LP + c4] = *(const float4*)&danger[(size_t)(b0 + r) * 256 + c4];
    }
    if (tid < 4 * 64) {                                      // sigs: 4 rows x 64 float4
        int s = tid >> 6, c4 = (tid & 63) * 4;
        *(float4*)&sSig[s * LP + c4] = *(const float4*)&sigs[(size_t)s * 256 + c4];
    }
    if (wave == 0) __builtin_amdgcn_s_wait_tensorcnt(0);
    __syncthreads();

    if (tid < 4) {  // 1/max(||sig_s||, 1e-12)
        float ss = 0.f;
        for (int d = 0; d < 256; ++d) { float v = sSig[tid*LP + d]; ss = fmaf(v, v, ss); }
        sIsg[tid] = 1.0f / fmaxf(sqrtf(ss), 1e-12f);
    }
    __syncthreads();

    // ---------------- Phase 1: per-row scalar reductions ----------------
    #pragma unroll 1
    for (int i = 0; i < 14; ++i) {
        int job = wave + 8*i;
        if (job < NER) {
            int r = job, g = r / 6;
            float aD = 0.f, a0 = 0.f, a1 = 0.f, a2 = 0.f, a3 = 0.f, an = 0.f;
            for (int d = lane; d < 256; d += 32) {
                float e = sE[r*LP + d];
                aD = fmaf(e, sDan[g*LP + d], aD);
                a0 = fmaf(e, sSig[0*LP + d], a0);
                a1 = fmaf(e, sSig[1*LP + d], a1);
                a2 = fmaf(e, sSig[2*LP + d], a2);
                a3 = fmaf(e, sSig[3*LP + d], a3);
                an = fmaf(e, e, an);
            }
            aD = wred32(aD); a0 = wred32(a0); a1 = wred32(a1);
            a2 = wred32(a2); a3 = wred32(a3); an = wred32(an);
            if (lane == 0) {
                float invE  = 1.0f / fmaxf(sqrtf(an), 1e-12f);
                float worst = fmaxf(fmaxf(a0*sIsg[0], a1*sIsg[1]),
                                    fmaxf(a2*sIsg[2], a3*sIsg[3])) * invE;
                sCo[r]  = aD;                               // co_stim
                sTol[r] = fmaxf(worst - 0.1f, 0.0f);        // relu(kappa*worst - eps_tol)
            }
        } else {
            int r = job - NER;
            float a0 = 0.f, a1 = 0.f, a2 = 0.f, a3 = 0.f, an = 0.f;
            for (int d = lane; d < 256; d += 32) {
                float k = sK[r*LP + d];
                a0 = fmaf(k, sSig[0*LP + d], a0);
                a1 = fmaf(k, sSig[1*LP + d], a1);
                a2 = fmaf(k, sSig[2*LP + d], a2);
                a3 = fmaf(k, sSig[3*LP + d], a3);
                an = fmaf(k, k, an);
            }
            a0 = wred32(a0); a1 = wred32(a1); a2 = wred32(a2);
            a3 = wred32(a3); an = wred32(an);
            if (lane == 0) {
                float invK = 1.0f / fmaxf(sqrtf(an), 1e-12f);
                sPss[r] = fmaxf(fmaxf(a0*sIsg[0], a1*sIsg[1]),
                                fmaxf(a2*sIsg[2], a3*sIsg[3])) * invK;
            }
        }
    }
    __syncthreads();

    // ---------------- Phase 2: LE = E@L^T, LK = K@L^T via v_wmma_f32_16x16x4_f32 ----------------
    const int rloc  = lane & 15;        // A: M row in tile; B: N col in tile
    const int hk    = (lane >> 4) * 2;  // half-wave K pair select
    #pragma unroll 1
    for (int i = 0; i < 14; ++i) {
        int job = wave + 8*i;
        const float* src; float* dst; int rt, j0;
        if (job < 48) { src = sE; dst = sLE; rt = (job >> 4) * 16; j0 = (job & 15) * 16; }
        else { int j = job - 48; src = sK; dst = sLK; rt = (j >> 4) * 16; j0 = (j & 15) * 16; }

        v8f acc = {0.f, 0.f, 0.f, 0.f, 0.f, 0.f, 0.f, 0.f};
        const float* arow = src + (rt + rloc) * LP + hk;
        const float* brow = Lmat + (size_t)(j0 + rloc) * 256 + hk;  // B[k][n] = L[n][k]
        #pragma unroll 4
        for (int k = 0; k < 256; k += 4) {
            v2f a = *(const v2f*)(arow + k);
            v2f b = *(const v2f*)(brow + k);
            acc = __builtin_amdgcn_wmma_f32_16x16x4_f32(
                      false, a, false, b, (short)0, acc, false, false);
        }
        int rhalf = (lane >> 4) * 8;
        #pragma unroll
        for (int v = 0; v < 8; ++v)
            dst[(rt + rhalf + v) * LP + j0 + rloc] = acc[v];
    }
    __syncthreads();

    // ---------------- Phase 3: dist = ||LE - LK||, cost matrix C ----------------
    #pragma unroll 1
    for (int i = 0; i < 48; ++i) {
        int p = wave + 8*i;                   // 384 pairs
        int g = p / 48, rem = p % 48, mm = rem >> 3, nn = rem & 7;
        int re = g*6 + mm, rk = g*8 + nn;
        float acc = 0.f;
        for (int d = lane; d < 256; d += 32) {
            float t = sLE[re*LP + d] - sLK[rk*LP + d];
            acc = fmaf(t, t, acc);
        }
        acc = wred32(acc);
        if (lane == 0) {
            float dist = sqrtf(fmaxf(acc, 1e-8f));
            sC[(g*6 + mm)*8 + nn] = dist - 0.9f*sCo[re] + 2.0f*sTol[re];
        }
    }
    __syncthreads();

    // ---------------- Phase 4: Sinkhorn + replicator + alpha + mix ----------------
    if (tid < BPW * 8) {
        const int g = tid >> 3, n = tid & 7;
        const int b = b0 + g;
        const float INV_EPS = 1.0f / (0.03f + 1e-8f);
        float lk[6];
        #pragma unroll
        for (int mm = 0; mm < 6; ++mm) lk[mm] = -sC[(g*6 + mm)*8 + n] * INV_EPS;

        const float log_u = logf(1.0f/6.0f + 1e-8f);
        const float log_v = logf(0.125f + 1e-8f);
        float la[6]; float lb = 0.f;
        #pragma unroll
        for (int mm = 0; mm < 6; ++mm) la[mm] = 0.f;

        #pragma unroll 1
        for (int it = 0; it < 30; ++it) {
            #pragma unroll
            for (int mm = 0; mm < 6; ++mm) {       // la = log_u - LSE_n(lk + lb)
                float x  = lk[mm] + lb;
                float mx = g8max(x);
                float s  = g8sum(__expf(x - mx));
                la[mm]   = log_u - (mx + __logf(s));
            }
            float mx = lk[0] + la[0];              // lb = log_v - LSE_m(lk + la)
            #pragma unroll
            for (int mm = 1; mm < 6; ++mm) mx = fmaxf(mx, lk[mm] + la[mm]);
            float s = 0.f;
            #pragma unroll
            for (int mm = 0; mm < 6; ++mm) s += __expf(lk[mm] + la[mm] - mx);
            lb = log_v - (mx + __logf(s));
        }
        float pmass = 0.f;
        #pragma unroll
        for (int mm = 0; mm < 6; ++mm)
            pmass += fminf(fmaxf(__expf(la[mm] + lk[mm] + lb), 0.f), 1.f);

        // replicator
        float wp  = w_prev[(size_t)b*8 + n];
        float fit = fitness[(size_t)b*8 + n];
        float base = g8sum(wp * fit);
        float crisis = crisis_level[b];
        if (!(crisis == crisis)) crisis = 0.f;           // nan_to_num
        float eta = 0.05f + 0.12f * crisis;
        float ltw = __logf(wp + 1e-8f) + eta * (fit - base) - 0.5f * sPss[g*8 + n];
        float z   = ltw * 2.5f;                          // / temp(0.4)
        float mz  = g8max(z);
        float ez  = __expf(z - mz);
        float tw  = ez / g8sum(ez);

        // crisis-adaptive alpha (decay is host-constant math, ~4e-9)
        float cc      = fminf(fmaxf(crisis, 0.f), 1.f);
        float alpha_c = 0.55f + (0.05f - 0.55f) * (1.0f - cosf(3.14159265358979323846f * cc)) * 0.5f;
        float dsv = delta_sharpe[b];
        if (!(dsv == dsv)) dsv = 0.f;
        float dt    = tanhf(dsv);
        float astar = fminf(fmaxf(alpha_c * (1.0f + 0.1f * dt), 0.05f), 0.55f);
        double span = 0.55 - 0.05 + 1e-9;
        double pos  = fmin(fmax((0.55 - 0.05) / span, 0.0), 1.0);
        double d2b  = fmin(pos, 1.0 - pos);
        double cen  = fmin(fmax(d2b * 2.0, 0.0), 1.0);
        float alpha_f = fminf(fmaxf(0.55f + 0.95f * (astar - 0.55f) * (float)cen, 0.05f), 0.55f);

        // mix + renormalize (twice, per reference)
        float w = (1.0f - alpha_f) * tw + alpha_f * pmass;
        if (!(w == w)) w = 0.125f;                       // nan -> 1/M
        float sw = g8sum(w);
        w = w / (sw + 1e-8f);
        w = fminf(fmaxf(w, 1e-6f), 1.0f);
        float sw2 = g8sum(w);
        out[(size_t)b*8 + n] = w / sw2;
    }
}

extern "C" void kernel_launch(void* const* d_in, const int* in_sizes, int n_in,
                              void* d_out, int out_size, void* d_ws, size_t ws_size,
                              hipStream_t stream) {
    const float* E      = (const float*)d_in[0];
    const float* K      = (const float*)d_in[1];
    const float* danger = (const float*)d_in[2];
    const float* w_prev = (const float*)d_in[3];
    const float* fit    = (const float*)d_in[4];
    const float* crisis = (const float*)d_in[5];
    const float* dsh    = (const float*)d_in[6];
    const float* Lmat   = (const float*)d_in[7];
    const float* sigs   = (const float*)d_in[8];
    float* out = (float*)d_out;

    const int B = in_sizes[3] / 8;      // w_prev is [B, 8]
    const int grid = B / BPW;           // 512 for B=4096

    (void)hipFuncSetAttribute((const void*)irt_fused,
                              hipFuncAttributeMaxDynamicSharedMemorySize, SMEM_BYTES);
    irt_fused<<<grid, 256, SMEM_BYTES, stream>>>(E, K, danger, w_prev, fit,
                                                 crisis, dsh, Lmat, sigs, out);
}